// RelPos_41901700940222
// MI455X (gfx1250) — compile-verified
//
#include <hip/hip_runtime.h>

typedef float v2f __attribute__((ext_vector_type(2)));
typedef float v4f __attribute__((ext_vector_type(4)));
typedef float v8f __attribute__((ext_vector_type(8)));

#define QL 28
#define KL 28
#define CH 32      // half of DIM
#define GDIM 64
#define NG 8

__global__ __launch_bounds__(256)
void relpos_wmma_kernel(const float* __restrict__ q,
                        const float* __restrict__ rel_h,
                        const float* __restrict__ rel_w,
                        float* __restrict__ out)
{
    __shared__ float sH[16][32];
    __shared__ float sW[16][32];

    const int blk = blockIdx.x;          // 0..3135
    const int bp  = blk / (QL * QL);     // batch pair 0..3
    const int ij  = blk % (QL * QL);
    const int qi  = ij / QL;
    const int qj  = ij % QL;

    const int tid  = threadIdx.x;
    const int wid  = tid >> 5;
    const int lane = tid & 31;

    if (wid < 2) {
        // Wave 0: lh (h-half channels, pos = qi); Wave 1: lw (w-half, pos = qj)
        const int half = lane >> 4;      // K-banding half
        const int m    = lane & 15;      // A row (within M tile) and B column
        const int bloc = m >> 3;         // local batch within pair
        const int g    = m & 7;          // group
        const int bb   = bp * 2 + bloc;
        const int coff = wid ? CH : 0;
        const float* emb = wid ? rel_w : rel_h;
        const int pos  = wid ? qj : qi;

        const float* qrow = q + ((size_t)(bb * (QL * QL) + ij)) * (NG * GDIM)
                              + g * GDIM + coff;

        // relative rows for the two N tiles; cols >= 28 are dead, clamp index
        int r0 = m - pos + 28;               // in [1, 43]
        int r1 = m + 16 - pos + 28;          // may exceed 55 only for dead cols
        if (r1 > 55) r1 = 55;
        const float* eb0 = emb + r0 * CH;
        const float* eb1 = emb + r1 * CH;

        v8f acc0 = {0.f, 0.f, 0.f, 0.f, 0.f, 0.f, 0.f, 0.f};
        v8f acc1 = {0.f, 0.f, 0.f, 0.f, 0.f, 0.f, 0.f, 0.f};

        #pragma unroll
        for (int kk = 0; kk < 8; ++kk) {
            const int c = kk * 4 + 2 * half;           // K offset per ISA layout
            v2f a  = { qrow[c], qrow[c + 1] };         // A[m, c..c+1]
            v2f b0 = { eb0[c],  eb0[c + 1]  };         // B[c..c+1, m]
            v2f b1 = { eb1[c],  eb1[c + 1]  };         // B[c..c+1, m+16]
            acc0 = __builtin_amdgcn_wmma_f32_16x16x4_f32(
                       false, a, false, b0, (short)0, acc0, false, false);
            acc1 = __builtin_amdgcn_wmma_f32_16x16x4_f32(
                       false, a, false, b1, (short)0, acc1, false, false);
        }

        // D layout: VGPR v -> row v + 8*half, col = lane%16 (+16 for tile 1)
        float (*S)[32] = wid ? sW : sH;
        #pragma unroll
        for (int v = 0; v < 8; ++v) {
            const int row = v + 8 * half;
            S[row][m] = acc0[v];
            if (m + 16 < 28) S[row][m + 16] = acc1[v];
        }
    }
    __syncthreads();

    // Streaming phase: 16 (b,g) rows x 784 elements, as 196 float4s per row.
    for (int idx = tid; idx < 16 * 196; idx += 256) {
        const int row = idx / 196;
        const int c4  = idx % 196;
        const int e0  = c4 * 4;
        const int bb  = bp * 2 + (row >> 3);
        const int g   = row & 7;
        v4f v;
        #pragma unroll
        for (int u = 0; u < 4; ++u) {
            const int e = e0 + u;
            const int k = e / 28;
            const int l = e - k * 28;
            v[u] = sH[row][k] + sW[row][l];
        }
        float* op = out + ((size_t)((bb * NG + g) * (QL * QL) + ij)) * (KL * KL) + e0;
        __builtin_nontemporal_store(v, (v4f*)op);
    }
}

extern "C" void kernel_launch(void* const* d_in, const int* in_sizes, int n_in,
                              void* d_out, int out_size, void* d_ws, size_t ws_size,
                              hipStream_t stream) {
    const float* q     = (const float*)d_in[0];
    const float* rel_h = (const float*)d_in[1];
    const float* rel_w = (const float*)d_in[2];
    // d_in[3] = onehot: its structure (k - i == r - 28) is applied analytically.
    (void)in_sizes; (void)n_in; (void)out_size; (void)d_ws; (void)ws_size;

    dim3 grid(4 * QL * QL);   // (batch pair) x (qi,qj) = 3136 blocks
    relpos_wmma_kernel<<<grid, 256, 0, stream>>>(q, rel_h, rel_w, (float*)d_out);
}